// MatrixNorm_9517647528373
// MI455X (gfx1250) — compile-verified
//
#include <hip/hip_runtime.h>
#include <cstdint>

// ---------------------------------------------------------------------------
// MatrixNorm (spectral norm via power iteration), MI455X / gfx1250.
//
// sigma = ||s_10|| / ||t_10|| with t = W*u, s = W^T*t fused in ONE pass over W
// per iteration (each W element consumed twice from LDS), 10 passes total:
// 2.56 GB HBM reads -> ~110 us at 23.3 TB/s. TDM streams row tiles into LDS,
// v_wmma_f32_16x16x4_f32 does the MACs, f32 atomics merge column partials.
// ---------------------------------------------------------------------------

#define NDIM      8192
#define ROWS      4                    // rows per block (tile = ROWS x NDIM)
#define RSTRIDE   (NDIM + 16)          // padded LDS row stride (bank-conflict free)
#define WAVES     16
#define BLOCKT    (WAVES * 32)         // 512 threads = 16 wave32
#define TILE_F    (ROWS * RSTRIDE)
#define COLS_PER_WAVE (NDIM / WAVES)   // 512
#define ITERS     10

typedef float v2f  __attribute__((ext_vector_type(2)));
typedef float v8f  __attribute__((ext_vector_type(8)));
typedef unsigned int u32x4 __attribute__((ext_vector_type(4)));
typedef int   i32x4 __attribute__((ext_vector_type(4)));
typedef int   i32x8 __attribute__((ext_vector_type(8)));

// --- Tensor Data Mover: DMA one 8192-float row of W into LDS ---------------
// D# group0: count=1, lds_addr, global byte addr, type=2 ("image").
// D# group1: data_size=4B, tensor_dim0=tensor_dim1=8192, tile=8192x1,
//            tensor_dim0_stride=8192. Tracked on TENSORcnt.
__device__ __forceinline__ void tdm_load_row(const float* gsrc, unsigned lds_byte_off) {
  unsigned long long ga = (unsigned long long)(uintptr_t)gsrc;
  u32x4 g0;
  g0[0] = 1u;                                            // count=1 (valid)
  g0[1] = lds_byte_off;                                  // lds_addr
  g0[2] = (unsigned)(ga & 0xFFFFFFFFull);                // global_addr[31:0]
  g0[3] = ((unsigned)(ga >> 32) & 0x01FFFFFFu)           // global_addr[56:32]
        | (2u << 30);                                    // type=2
  i32x8 g1;
  g1[0] = (int)(2u << 16);                               // data_size: 4 bytes
  g1[1] = (int)((NDIM & 0xFFFF) << 16);                  // tensor_dim0 lo16 @bits63:48
  g1[2] = (int)(((NDIM >> 16) & 0xFFFF)                  // tensor_dim0 hi16
        | ((NDIM & 0xFFFF) << 16));                      // tensor_dim1 lo16
  g1[3] = (int)(((NDIM >> 16) & 0xFFFF)                  // tensor_dim1 hi16
        | (NDIM << 16));                                 // tile_dim0 = 8192
  g1[4] = 1;                                             // tile_dim1 = 1 row
  g1[5] = NDIM;                                          // tensor_dim0_stride lo32
  g1[6] = 0;
  g1[7] = 0;
  i32x4 gz = {0, 0, 0, 0};
#if defined(__clang_major__) && (__clang_major__ >= 23)
  i32x8 gz8 = {0, 0, 0, 0, 0, 0, 0, 0};
  __builtin_amdgcn_tensor_load_to_lds(g0, g1, gz, gz, gz8, 0);
#else
  __builtin_amdgcn_tensor_load_to_lds(g0, g1, gz, gz, 0);
#endif
}

// --- Fused kernel: t = W_tile * u ; s += t^T * W_tile ; ||t||^2 += ---------
__global__ __launch_bounds__(BLOCKT)
void MatrixNorm_fused(const float* __restrict__ W,
                      const float* __restrict__ u,
                      float* __restrict__ s,
                      float* __restrict__ tnorm2) {
  extern __shared__ float smem[];              // [TILE_F | red 64 | t 4]
  const int tid  = threadIdx.x;
  const int wave = tid >> 5;
  const int lane = tid & 31;
  const int half = lane >> 4;                  // 0: lanes 0-15, 1: lanes 16-31
  const long long row0 = (long long)blockIdx.x * ROWS;

  // Wave 0 streams the 4-row tile into LDS via TDM, then releases the block.
  if (tid < 32) {
    for (int r = 0; r < ROWS; ++r)
      tdm_load_row(W + (row0 + r) * (long long)NDIM, (unsigned)(r * RSTRIDE * 4u));
    __builtin_amdgcn_s_wait_tensorcnt(0);
  }
  __syncthreads();

  const int cbase = wave * COLS_PER_WAVE;
  float* red = smem + TILE_F;                  // 16 waves x 4 partial dots
  float* tsh = smem + TILE_F + 64;             // final t[0..3]

  // ---- Phase 1: t[r] = dot(Wtile[r,:], u), WMMA 16x16x4 f32.
  // A: lane l holds Wtile[(l&3)][c+2h .. +1]  (rows replicated 4x across M)
  // B: u[c+2h .. +1] broadcast over the 16 output columns -> every output
  //    column of D equals the row dot; lane 0 ends up with t-partials in C[0..3].
  {
    const float* arow_p = smem + (lane & 3) * RSTRIDE;
    v8f acc0 = {0.f,0.f,0.f,0.f,0.f,0.f,0.f,0.f};
    v8f acc1 = {0.f,0.f,0.f,0.f,0.f,0.f,0.f,0.f};
    #pragma unroll 4
    for (int c = cbase; c < cbase + COLS_PER_WAVE; c += 8) {
      const int cc0 = c + 2 * half;
      const int cc1 = c + 4 + 2 * half;
      v2f a0 = *(const v2f*)(arow_p + cc0);    // ds_load_b64 (padded: no bank conflict)
      v2f b0 = *(const v2f*)(u + cc0);         // global b64, L2/WGP$ broadcast
      v2f a1 = *(const v2f*)(arow_p + cc1);
      v2f b1 = *(const v2f*)(u + cc1);
      acc0 = __builtin_amdgcn_wmma_f32_16x16x4_f32(false, a0, false, b0,
                                                   (short)0, acc0, false, false);
      acc1 = __builtin_amdgcn_wmma_f32_16x16x4_f32(false, a1, false, b1,
                                                   (short)0, acc1, false, false);
    }
    v8f acc = acc0 + acc1;
    if (lane == 0) {
      red[wave * 4 + 0] = acc[0];
      red[wave * 4 + 1] = acc[1];
      red[wave * 4 + 2] = acc[2];
      red[wave * 4 + 3] = acc[3];
    }
  }
  __syncthreads();
  if (tid < 4) {
    float t = 0.f;
    for (int w2 = 0; w2 < WAVES; ++w2) t += red[w2 * 4 + tid];
    tsh[tid] = t;
  }
  __syncthreads();
  if (tid == 0) {
    float t0 = tsh[0], t1 = tsh[1], t2 = tsh[2], t3 = tsh[3];
    atomicAdd(tnorm2, t0 * t0 + t1 * t1 + t2 * t2 + t3 * t3);
  }

  // ---- Phase 2: s[n] += sum_r t[r]*Wtile[r][n], WMMA 16x16x4 f32.
  // A: t replicated across all 16 M rows (pure register broadcast, no loads).
  // B: Wtile[k][c + (lane&15)] -> 64 LDS reads per WMMA, all useful.
  // D rows are identical; lanes 0-15 commit D[0][n] with one f32 atomic.
  {
    v2f ta; ta.x = tsh[2 * half]; ta.y = tsh[2 * half + 1];
    const float* rB0 = smem + (2 * half + 0) * RSTRIDE;
    const float* rB1 = smem + (2 * half + 1) * RSTRIDE;
    const v8f zc = {0.f,0.f,0.f,0.f,0.f,0.f,0.f,0.f};
    #pragma unroll 2
    for (int c = cbase; c < cbase + COLS_PER_WAVE; c += 16) {
      const int n = c + (lane & 15);
      v2f b; b.x = rB0[n]; b.y = rB1[n];
      v8f d = __builtin_amdgcn_wmma_f32_16x16x4_f32(false, ta, false, b,
                                                    (short)0, zc, false, false);
      if (lane < 16) atomicAdd(&s[n], d[0]);
    }
  }
}

// --- init: u_buf = u_in, s = 0, ||t||^2 = 0 --------------------------------
__global__ void MatrixNorm_init(const float* __restrict__ uin,
                                float* __restrict__ u,
                                float* __restrict__ s,
                                float* __restrict__ tnorm2) {
  int i = blockIdx.x * blockDim.x + threadIdx.x;
  if (i < NDIM) { u[i] = uin[i]; s[i] = 0.f; }
  if (i == 0) *tnorm2 = 0.f;
}

// --- normalize: u = s/||s||; clear accumulators; last iter -> sigma --------
__global__ __launch_bounds__(1024)
void MatrixNorm_normalize(float* __restrict__ s,
                          float* __restrict__ u,
                          float* __restrict__ tnorm2,
                          float* __restrict__ out,
                          int last) {
  __shared__ float red[32];
  const int tid = threadIdx.x;                 // 1024 threads = 32 wave32
  float acc = 0.f;
  for (int i = tid; i < NDIM; i += 1024) { float v = s[i]; acc += v * v; }
  for (int off = 16; off > 0; off >>= 1) acc += __shfl_down(acc, off, 32);
  if ((tid & 31) == 0) red[tid >> 5] = acc;
  __syncthreads();
  if (tid < 32) {
    float v = red[tid];
    for (int off = 16; off > 0; off >>= 1) v += __shfl_down(v, off, 32);
    if (tid == 0) red[0] = v;
  }
  __syncthreads();
  const float sn2 = red[0];
  const float inv = rsqrtf(sn2);
  for (int i = tid; i < NDIM; i += 1024) { u[i] = s[i] * inv; s[i] = 0.f; }
  if (tid == 0) {
    if (last) out[0] = sqrtf(sn2 / *tnorm2);   // sigma = ||s|| / ||t||
    *tnorm2 = 0.f;
  }
}

extern "C" void kernel_launch(void* const* d_in, const int* in_sizes, int n_in,
                              void* d_out, int out_size, void* d_ws, size_t ws_size,
                              hipStream_t stream) {
  const float* W  = (const float*)d_in[0];   // [8192, 8192] fp32
  const float* u0 = (const float*)d_in[1];   // [1, 8192] fp32, unit norm
  float* out = (float*)d_out;                // [1] fp32

  float* s   = (float*)d_ws;                 // 8192: s accumulator
  float* ub  = s + NDIM;                     // 8192: current unit vector
  float* tn  = ub + NDIM;                    // 1:    ||t||^2 accumulator

  MatrixNorm_init<<<(NDIM + 255) / 256, 256, 0, stream>>>(u0, ub, s, tn);

  const size_t lds_bytes = (size_t)(TILE_F + 64 + 4) * sizeof(float); // ~128.5 KB
  for (int it = 0; it < ITERS; ++it) {
    MatrixNorm_fused<<<NDIM / ROWS, BLOCKT, lds_bytes, stream>>>(W, ub, s, tn);
    MatrixNorm_normalize<<<1, 1024, 0, stream>>>(s, ub, tn, out, it == ITERS - 1);
  }
}